// ra_cluster_90950227460804
// MI455X (gfx1250) — compile-verified
//
#include <hip/hip_runtime.h>

#define CANVAS_X   1024
#define CANVAS_PIX (1024 * 1024)
#define N_RAW_C    1200000
#define N_SEL_C    1000000
#define N_RA_C     200000

typedef int   v4i __attribute__((ext_vector_type(4)));
typedef float v4f __attribute__((ext_vector_type(4)));

// CDNA5 single-instruction float max atomic, device scope (RMW at the L2
// atomic units, coherent across WGPs). Non-returning -> tracked with STOREcnt.
__device__ __forceinline__ void global_atomic_max_f32_dev(float* addr, float val) {
    asm volatile("global_atomic_max_num_f32 %0, %1, off scope:SCOPE_DEV"
                 :
                 : "v"(addr), "v"(val)
                 : "memory");
}

// ---------------------------------------------------------------------------
// Kernel 1: canvas = 0.0f. Labels are >= 0, so 0-init + maxNum is exactly
// equivalent to the reference's (-inf init, scatter-max, isfinite?v:0).
// 16B per thread via b128 stores.
// ---------------------------------------------------------------------------
__global__ void ra_canvas_init_kernel(v4f* __restrict__ canvas, int n4) {
    int i = blockIdx.x * blockDim.x + threadIdx.x;
    if (i < n4) {
        canvas[i] = (v4f){0.0f, 0.0f, 0.0f, 0.0f};
    }
}

// ---------------------------------------------------------------------------
// Kernel 2: scatter-max labels onto the per-batch canvas.
// 4 points per thread: 1x int4 valid + 3x int4 coor (all 16B aligned since
// the coor group base is 48*tid bytes), NT-hinted (read-once streams so the
// L2-resident canvas is not evicted). 4 label gathers (random within an
// L2-resident 4.8MB slab -> default RT policy) + 4 fmax atomics.
// ---------------------------------------------------------------------------
__global__ void ra_scatter_max_kernel(const float* __restrict__ lidar,  // [B, N_RAW]
                                      const int*   __restrict__ valid,  // [B, N_SEL]
                                      const int*   __restrict__ coor,   // [B, N_SEL, 3]
                                      float*       __restrict__ canvas, // [B, CANVAS_PIX]
                                      int ngroups)                      // B * N_SEL / 4
{
    int t = blockIdx.x * blockDim.x + threadIdx.x;
    if (t >= ngroups) return;

    int p = t * 4;                 // first point of this group
    int b = p / N_SEL_C;           // N_SEL_C % 4 == 0 -> whole group in batch b

    v4i vv = __builtin_nontemporal_load((const v4i*)(valid + p));

    const v4i* cb = (const v4i*)(coor + 3LL * (long long)p);  // 48*t bytes: aligned
    v4i c0 = __builtin_nontemporal_load(cb + 0);
    v4i c1 = __builtin_nontemporal_load(cb + 1);
    v4i c2 = __builtin_nontemporal_load(cb + 2);

    // (y, x) per point within the 12-int group:
    int y0 = c0.y, x0 = c0.z;      // idx 1, 2
    int y1 = c1.x, x1 = c1.y;      // idx 4, 5
    int y2 = c1.w, x2 = c2.x;      // idx 7, 8
    int y3 = c2.z, x3 = c2.w;      // idx 10, 11

    const float* lb = lidar + (long long)b * N_RAW_C;
    float l0 = lb[vv.x];
    float l1 = lb[vv.y];
    float l2 = lb[vv.z];
    float l3 = lb[vv.w];

    float* cv = canvas + (long long)b * CANVAS_PIX;
    global_atomic_max_f32_dev(cv + (y0 * CANVAS_X + x0), l0);
    global_atomic_max_f32_dev(cv + (y1 * CANVAS_X + x1), l1);
    global_atomic_max_f32_dev(cv + (y2 * CANVAS_X + x2), l2);
    global_atomic_max_f32_dev(cv + (y3 * CANVAS_X + x3), l3);
}

// ---------------------------------------------------------------------------
// Kernel 3: out[b][j] = canvas[b][radar_ind[b][j]]   (canvas hits L2)
// 4 outputs per thread: int4 index load, 4 scalar gathers, float4 NT store.
// ---------------------------------------------------------------------------
__global__ void ra_gather_kernel(const float* __restrict__ canvas, // [B, CANVAS_PIX]
                                 const int*   __restrict__ rind,   // [B, N_RA]
                                 float*       __restrict__ out,    // [B, N_RA]
                                 int ngroups)                      // B * N_RA / 4
{
    int t = blockIdx.x * blockDim.x + threadIdx.x;
    if (t >= ngroups) return;

    int p = t * 4;
    int b = p / N_RA_C;            // N_RA_C % 4 == 0 -> whole group in batch b

    v4i idx = __builtin_nontemporal_load((const v4i*)(rind + p));

    const float* cv = canvas + (long long)b * CANVAS_PIX;
    v4f r;
    r.x = cv[idx.x];
    r.y = cv[idx.y];
    r.z = cv[idx.z];
    r.w = cv[idx.w];

    __builtin_nontemporal_store(r, (v4f*)(out + p));
}

extern "C" void kernel_launch(void* const* d_in, const int* in_sizes, int n_in,
                              void* d_out, int out_size, void* d_ws, size_t ws_size,
                              hipStream_t stream) {
    const float* lidar = (const float*)d_in[0]; // [B, N_RAW] float32
    const int*   valid = (const int*)d_in[1];   // [B, N_SEL] int
    const int*   coor  = (const int*)d_in[2];   // [B, N_SEL, 3] int
    const int*   rind  = (const int*)d_in[3];   // [B, N_RA] int
    float*       out   = (float*)d_out;         // [B, N_RA] float32

    const int B = in_sizes[1] / N_SEL_C;
    float* canvas = (float*)d_ws;               // B * CANVAS_PIX floats (16 MB @ B=4)

    const int threads = 256;

    // 1) zero canvas (b128 stores)
    {
        int n4 = (B * CANVAS_PIX) / 4;
        int blocks = (n4 + threads - 1) / threads;
        ra_canvas_init_kernel<<<blocks, threads, 0, stream>>>((v4f*)canvas, n4);
    }

    // 2) scatter-max (4 points / thread)
    {
        int ngroups = (B * N_SEL_C) / 4;
        int blocks = (ngroups + threads - 1) / threads;
        ra_scatter_max_kernel<<<blocks, threads, 0, stream>>>(lidar, valid, coor,
                                                              canvas, ngroups);
    }

    // 3) gather at radar indices (4 outputs / thread)
    {
        int ngroups = (B * N_RA_C) / 4;
        int blocks = (ngroups + threads - 1) / threads;
        ra_gather_kernel<<<blocks, threads, 0, stream>>>(canvas, rind, out, ngroups);
    }
}